// MultiView2DProjectionLoss_24206435680919
// MI455X (gfx1250) — compile-verified
//
#include <hip/hip_runtime.h>
#include <hip/hip_bf16.h>

#define NUM_OBJECTS 80000
#define EPS_W 1e-05f
#define EPS_A 1e-07f

typedef __attribute__((ext_vector_type(2))) float v2f;
typedef __attribute__((ext_vector_type(8))) float v8f;

// Corner sign tables (match X_SIGN / Y_SIGN / Z_SIGN in the reference)
__constant__ float XS[8] = { 1.f, 1.f, -1.f, -1.f, 1.f, 1.f, -1.f, -1.f };
__constant__ float YS[8] = { 1.f, -1.f, -1.f, 1.f, 1.f, -1.f, -1.f, 1.f };
__constant__ float ZS[8] = { 1.f, 1.f, 1.f, 1.f, -1.f, -1.f, -1.f, -1.f };

// ---------------------------------------------------------------------------
// Kernel 0: zero the workspace (sums[80000], counts[80000], accum[2]).
// d_ws is poisoned with 0xAA by the harness, so we must clear it ourselves.
// ---------------------------------------------------------------------------
__global__ void k0_zero_ws(float* ws, int n) {
    int i = blockIdx.x * blockDim.x + threadIdx.x;
    if (i < n) ws[i] = 0.0f;
}

// ---------------------------------------------------------------------------
// Kernel 1: one thread per box. HBM-bandwidth-bound streaming pass:
//   corners -> per-box 4x4 projection (rows 0..2) -> min/max -> clip ->
//   GIoU loss -> atomic scatter into per-object sum/count (L2-resident).
// ---------------------------------------------------------------------------
__global__ void k1_box_loss(const float* __restrict__ pred,   // N x 7
                            const float* __restrict__ gt,     // N x 4
                            const float* __restrict__ l2i,    // N x 4 x 4
                            const float* __restrict__ shp,    // N x 2
                            const int*   __restrict__ ids,    // N
                            float* __restrict__ sums,
                            float* __restrict__ cnts,
                            int n) {
    int i = blockIdx.x * blockDim.x + threadIdx.x;
    if (i >= n) return;

    // ---- loads (coalesced; lidar2img rows are 16B-aligned float4) ----
    const float* p = pred + (size_t)i * 7;
    float bx = p[0], by = p[1], bz = p[2];
    float hl = 0.5f * p[3], hw = 0.5f * p[4], hh = 0.5f * p[5];
    float yaw = p[6];

    const float4* M = (const float4*)(l2i + (size_t)i * 16);
    float4 m0 = M[0];   // row 0
    float4 m1 = M[1];   // row 1
    float4 m2 = M[2];   // row 2  (row 3 unused)

    float4 g = ((const float4*)gt)[i];
    float2 ishp = ((const float2*)shp)[i];
    float Himg = ishp.x, Wimg = ishp.y;

    // precise sin/cos to track the reference (jnp.cos / jnp.sin)
    float s = sinf(yaw);
    float c = cosf(yaw);

    float minx =  3.4e38f, maxx = -3.4e38f;
    float miny =  3.4e38f, maxy = -3.4e38f;

#pragma unroll
    for (int k = 0; k < 8; ++k) {
        float xc = hl * XS[k];
        float yc = hw * YS[k];
        float zc = hh * ZS[k];
        // R(yaw) * corner + center
        float px = fmaf(c, xc, fmaf(-s, yc, bx));
        float py = fmaf(s, xc, fmaf( c, yc, by));
        float pz = zc + bz;
        // rows 0..2 of lidar2img applied to (px,py,pz,1)
        float hx  = fmaf(m0.x, px, fmaf(m0.y, py, fmaf(m0.z, pz, m0.w)));
        float hy  = fmaf(m1.x, px, fmaf(m1.y, py, fmaf(m1.z, pz, m1.w)));
        float hwc = fmaf(m2.x, px, fmaf(m2.y, py, fmaf(m2.z, pz, m2.w)));
        float wc = fmaxf(hwc, EPS_W);
        float xi = hx / wc;
        float yi = hy / wc;
        minx = fminf(minx, xi); maxx = fmaxf(maxx, xi);
        miny = fminf(miny, yi); maxy = fmaxf(maxy, yi);
    }

    // clip to [0, W] x [0, H]
    float p0 = fminf(fmaxf(minx, 0.0f), Wimg);
    float p1 = fminf(fmaxf(miny, 0.0f), Himg);
    float p2 = fminf(fmaxf(maxx, 0.0f), Wimg);
    float p3 = fminf(fmaxf(maxy, 0.0f), Himg);

    // GIoU
    float x1 = fmaxf(p0, g.x), y1 = fmaxf(p1, g.y);
    float x2 = fminf(p2, g.z), y2 = fminf(p3, g.w);
    float inter = fmaxf(x2 - x1, 0.0f) * fmaxf(y2 - y1, 0.0f);
    float pa = (p2 - p0) * (p3 - p1);
    float ga = (g.z - g.x) * (g.w - g.y);
    float uni = pa + ga - inter + EPS_A;
    float iou = inter / uni;
    float cx1 = fminf(p0, g.x), cy1 = fminf(p1, g.y);
    float cx2 = fmaxf(p2, g.z), cy2 = fmaxf(p3, g.w);
    float carea = fmaxf(cx2 - cx1, 0.0f) * fmaxf(cy2 - cy1, 0.0f) + EPS_A;
    float giou = iou - (carea - uni) / carea;
    float loss = 1.0f - giou;

    int id = ids[i];
    atomicAdd(&sums[id], loss);
    atomicAdd(&cnts[id], 1.0f);
}

// ---------------------------------------------------------------------------
// Kernel 2: segment finalize + reduce via V_WMMA_F32_16X16X4_F32.
// With B == all-ones, D = A*B + C accumulates row-sums of A into every
// column of D: each WMMA folds 64 segment values per wave. Two accumulator
// chains: one for segment means, one for presence flags. Final total is
// lanesum(l) + lanesum(l XOR 16) (D layout: lane<16 holds M=0..7,
// lane>=16 holds M=8..15; all columns identical).
// ---------------------------------------------------------------------------
__global__ void k2_reduce_wmma(const float* __restrict__ sums,
                               const float* __restrict__ cnts,
                               float* __restrict__ accum) {
    const int lane   = threadIdx.x & 31;
    const int gwave  = (blockIdx.x * blockDim.x + threadIdx.x) >> 5;
    const int nwaves = (gridDim.x * blockDim.x) >> 5;

    v2f bones; bones[0] = 1.0f; bones[1] = 1.0f;
    v8f dmean = {};
    v8f dpres = {};

    // wave-uniform grid-stride loop: 64 segments per wave per iteration
    for (int base = gwave * 64; base < NUM_OBJECTS; base += nwaves * 64) {
        int j0 = base + 2 * lane;
        int j1 = j0 + 1;
        float c0 = (j0 < NUM_OBJECTS) ? cnts[j0] : 0.0f;
        float s0 = (j0 < NUM_OBJECTS) ? sums[j0] : 0.0f;
        float c1 = (j1 < NUM_OBJECTS) ? cnts[j1] : 0.0f;
        float s1 = (j1 < NUM_OBJECTS) ? sums[j1] : 0.0f;

        v2f am, ap;
        am[0] = (c0 > 0.0f) ? s0 / c0 : 0.0f;   // seg_mean (masked by present)
        am[1] = (c1 > 0.0f) ? s1 / c1 : 0.0f;
        ap[0] = (c0 > 0.0f) ? 1.0f : 0.0f;      // present flag
        ap[1] = (c1 > 0.0f) ? 1.0f : 0.0f;

        dmean = __builtin_amdgcn_wmma_f32_16x16x4_f32(
            false, am, false, bones, (short)0, dmean, false, false);
        dpres = __builtin_amdgcn_wmma_f32_16x16x4_f32(
            false, ap, false, bones, (short)0, dpres, false, false);
    }

    float ps = dmean[0] + dmean[1] + dmean[2] + dmean[3] +
               dmean[4] + dmean[5] + dmean[6] + dmean[7];
    float pc = dpres[0] + dpres[1] + dpres[2] + dpres[3] +
               dpres[4] + dpres[5] + dpres[6] + dpres[7];
    ps += __shfl_xor(ps, 16, 32);
    pc += __shfl_xor(pc, 16, 32);

    if (lane == 0) {
        atomicAdd(&accum[0], ps);
        atomicAdd(&accum[1], pc);
    }
}

// ---------------------------------------------------------------------------
// Kernel 3: final scalar (LOSS_WEIGHT == 1.0)
// ---------------------------------------------------------------------------
__global__ void k3_finalize(const float* __restrict__ accum,
                            float* __restrict__ out) {
    if (blockIdx.x == 0 && threadIdx.x == 0) {
        out[0] = accum[0] / accum[1];
    }
}

extern "C" void kernel_launch(void* const* d_in, const int* in_sizes, int n_in,
                              void* d_out, int out_size, void* d_ws, size_t ws_size,
                              hipStream_t stream) {
    const float* pred = (const float*)d_in[0];   // N x 7
    const float* gt   = (const float*)d_in[1];   // N x 4
    const float* l2i  = (const float*)d_in[2];   // N x 16
    const float* shp  = (const float*)d_in[3];   // N x 2
    const int*   ids  = (const int*)d_in[4];     // N
    float* out = (float*)d_out;

    const int N = in_sizes[0] / 7;

    float* sums  = (float*)d_ws;                 // NUM_OBJECTS
    float* cnts  = sums + NUM_OBJECTS;           // NUM_OBJECTS
    float* accum = cnts + NUM_OBJECTS;           // 2

    // 0) zero sums/counts/accum
    int nz = 2 * NUM_OBJECTS + 2;
    k0_zero_ws<<<(nz + 255) / 256, 256, 0, stream>>>((float*)d_ws, nz);

    // 1) per-box loss + scatter
    k1_box_loss<<<(N + 255) / 256, 256, 0, stream>>>(
        pred, gt, l2i, shp, ids, sums, cnts, N);

    // 2) WMMA segment reduce
    k2_reduce_wmma<<<64, 256, 0, stream>>>(sums, cnts, accum);

    // 3) final divide
    k3_finalize<<<1, 32, 0, stream>>>(accum, out);
}